// FasterRCNN_68040871903296
// MI455X (gfx1250) — compile-verified
//
#include <hip/hip_runtime.h>
#include <hip/hip_bf16.h>
#include <math.h>

#define MG    48
#define CCH   256
#define NCELL 2304      // 48*48
#define NA    6912      // anchors
#define KA    3
#define RPOST 512
#define HID   1024
#define FIN   2304      // 9*256
#define IMGF  1536.0f
#define NMS_THR 0.3f
#define SORTN 8192

// FC1 tiling
#define BM 64
#define BN 128
#define BK 32
#define ASTR 34         // A LDS row stride (even -> 8B-aligned b64, bank-permuting)
#define BSTR 136        // B LDS row stride (K and K+2 rows land in disjoint bank halves)

typedef __attribute__((ext_vector_type(2)))  float  v2f;
typedef __attribute__((ext_vector_type(8)))  float  v8f;

// ---------------------------------------------------------------------------
// 1) RPN head: per grid cell, 18 dots of length 256 (6 obj logits, 12 regs)
// ---------------------------------------------------------------------------
__global__ void rpn_head_kernel(const float* __restrict__ feat,
                                const float* __restrict__ Wobj,
                                const float* __restrict__ bobj,
                                const float* __restrict__ Wreg,
                                const float* __restrict__ breg,
                                float* __restrict__ rpn_out) {
    __shared__ float frow[CCH];
    const int cell = blockIdx.x;
    const int tid  = threadIdx.x;
    frow[tid] = feat[cell * CCH + tid];
    __syncthreads();
    if (tid < 18) {
        float s;
        if (tid < 6) {
            s = bobj[tid];
            for (int c = 0; c < CCH; ++c) s = fmaf(frow[c], Wobj[c * 6 + tid], s);
        } else {
            const int j = tid - 6;
            s = breg[j];
            for (int c = 0; c < CCH; ++c) s = fmaf(frow[c], Wreg[c * 12 + j], s);
        }
        rpn_out[cell * 18 + tid] = s;
    }
}

// ---------------------------------------------------------------------------
// 2) Anchor decode: softmax over 2 logits, unparameterize, clip
// ---------------------------------------------------------------------------
__global__ void anchor_decode_kernel(const float* __restrict__ rpn_out,
                                     const float* __restrict__ anchors,
                                     float* __restrict__ scores,
                                     float* __restrict__ boxes,
                                     float* __restrict__ probs) {
    const int a = blockIdx.x * blockDim.x + threadIdx.x;
    if (a >= NA) return;
    const int cell = a / KA;
    const int k    = a % KA;
    const float l0 = rpn_out[cell * 18 + 2 * k];
    const float l1 = rpn_out[cell * 18 + 2 * k + 1];
    const float mx = fmaxf(l0, l1);
    const float e0 = expf(l0 - mx), e1 = expf(l1 - mx);
    const float inv = 1.0f / (e0 + e1);
    const float p0 = e0 * inv, p1 = e1 * inv;
    probs[a * 2 + 0] = p0;
    probs[a * 2 + 1] = p1;
    scores[a] = p1;

    const float ax = anchors[a * 4 + 0], ay = anchors[a * 4 + 1];
    const float aw = anchors[a * 4 + 2], ah = anchors[a * 4 + 3];
    const float r0 = rpn_out[cell * 18 + 6 + 4 * k + 0];
    const float r1 = rpn_out[cell * 18 + 6 + 4 * k + 1];
    const float r2 = rpn_out[cell * 18 + 6 + 4 * k + 2];
    const float r3 = rpn_out[cell * 18 + 6 + 4 * k + 3];
    const float cxa = ax + aw * 0.5f, cya = ay + ah * 0.5f;
    const float cx = r0 * aw + cxa, cy = r1 * ah + cya;
    const float w  = aw * expf(r2), hh = ah * expf(r3);
    const float x1u = cx - w * 0.5f, y1u = cy - hh * 0.5f;
    const float X1 = fminf(fmaxf(x1u, 0.0f), IMGF);
    const float Y1 = fminf(fmaxf(y1u, 0.0f), IMGF);
    const float X2 = fminf(fmaxf(x1u + w, 0.0f), IMGF);
    const float Y2 = fminf(fmaxf(y1u + hh, 0.0f), IMGF);
    boxes[a * 4 + 0] = X1;
    boxes[a * 4 + 1] = Y1;
    boxes[a * 4 + 2] = X2 - X1;
    boxes[a * 4 + 3] = Y2 - Y1;
}

// ---------------------------------------------------------------------------
// 3) Descending argsort of 6912 scores: in-LDS bitonic on 8192 slots (64 KB)
// ---------------------------------------------------------------------------
__global__ void sort_kernel(const float* __restrict__ scores,
                            int* __restrict__ sorted_idx) {
    __shared__ float key[SORTN];
    __shared__ int   idx[SORTN];
    const int tid = threadIdx.x;
    for (int i = tid; i < SORTN; i += blockDim.x) {
        key[i] = (i < NA) ? -scores[i] : 3.0e38f;   // ascending on -score
        idx[i] = i;
    }
    __syncthreads();
    for (unsigned k = 2; k <= SORTN; k <<= 1) {
        for (unsigned j = k >> 1; j > 0; j >>= 1) {
            for (unsigned p = tid; p < SORTN; p += blockDim.x) {
                const unsigned q = p ^ j;
                if (q > p) {
                    const bool up = ((p & k) == 0);
                    if ((key[p] > key[q]) == up) {
                        float tk = key[p]; key[p] = key[q]; key[q] = tk;
                        int   ti = idx[p]; idx[p] = idx[q]; idx[q] = ti;
                    }
                }
            }
            __syncthreads();
        }
    }
    for (int i = tid; i < NA; i += blockDim.x) sorted_idx[i] = idx[i];
}

// ---------------------------------------------------------------------------
// 4) Greedy NMS over sorted boxes (serial scan, boxes staged in dynamic LDS)
// ---------------------------------------------------------------------------
__global__ void nms_kernel(const float* __restrict__ boxes,
                           const int* __restrict__ sorted_idx,
                           int* __restrict__ keep_out) {
    extern __shared__ unsigned char smemraw[];
    float* x1 = (float*)smemraw;
    float* y1 = x1 + NA;
    float* x2 = y1 + NA;
    float* y2 = x2 + NA;
    unsigned char* keep = (unsigned char*)(y2 + NA);
    __shared__ int s_sup;
    const int tid = threadIdx.x;

    for (int i = tid; i < NA; i += blockDim.x) {
        const int a = sorted_idx[i];
        const float bx = boxes[a * 4 + 0], by = boxes[a * 4 + 1];
        const float bw = boxes[a * 4 + 2], bh = boxes[a * 4 + 3];
        x1[i] = bx; y1[i] = by; x2[i] = bx + bw; y2[i] = by + bh;
        keep[i] = 0;
    }
    __syncthreads();

    for (int i = 0; i < NA; ++i) {
        if (tid == 0) s_sup = 0;
        __syncthreads();
        const float ix1 = x1[i], iy1 = y1[i], ix2 = x2[i], iy2 = y2[i];
        const float iarea = (ix2 - ix1) * (iy2 - iy1);
        int flag = 0;
        for (int j = tid; j < i; j += blockDim.x) {
            if (keep[j]) {
                const float iw = fmaxf(0.0f, fminf(ix2, x2[j]) - fmaxf(ix1, x1[j]));
                const float ih = fmaxf(0.0f, fminf(iy2, y2[j]) - fmaxf(iy1, y1[j]));
                const float inter = iw * ih;
                const float jarea = (x2[j] - x1[j]) * (y2[j] - y1[j]);
                const float iou = inter / (iarea + jarea - inter + 1e-8f);
                if (iou > NMS_THR) flag = 1;
            }
        }
        if (flag) atomicOr(&s_sup, 1);
        __syncthreads();
        if (tid == 0) keep[i] = (s_sup == 0) ? 1 : 0;
        __syncthreads();
    }
    for (int i = tid; i < NA; i += blockDim.x) keep_out[i] = keep[i];
}

// ---------------------------------------------------------------------------
// 5) Selection (top-512 kept first, suppressed as filler) + emit cols 0..5
// ---------------------------------------------------------------------------
__global__ void select_kernel(const int* __restrict__ sorted_idx,
                              const int* __restrict__ keep,
                              const float* __restrict__ probs,
                              const float* __restrict__ boxes,
                              int* __restrict__ sel,
                              float* __restrict__ out) {
    __shared__ int sel_s[RPOST];
    if (threadIdx.x == 0) {
        int cnt = 0;
        for (int i = 0; i < NA && cnt < RPOST; ++i)
            if (keep[i]) sel_s[cnt++] = sorted_idx[i];
        for (int i = 0; i < NA && cnt < RPOST; ++i)
            if (!keep[i]) sel_s[cnt++] = sorted_idx[i];
    }
    __syncthreads();
    const int r = threadIdx.x;
    if (r < RPOST) {
        const int a = sel_s[r];
        sel[r] = a;
        out[r * 10 + 0] = probs[a * 2 + 0];
        out[r * 10 + 1] = probs[a * 2 + 1];
        out[r * 10 + 2] = boxes[a * 4 + 0];
        out[r * 10 + 3] = boxes[a * 4 + 1];
        out[r * 10 + 4] = boxes[a * 4 + 2];
        out[r * 10 + 5] = boxes[a * 4 + 3];
    }
}

// ---------------------------------------------------------------------------
// 6) ROIAlign: one block per ROI, one thread per channel (coalesced over C)
// ---------------------------------------------------------------------------
__global__ void roialign_kernel(const float* __restrict__ feat,
                                const float* __restrict__ out, // reads cols 2..5
                                float* __restrict__ flat) {
    const int r = blockIdx.x;
    const int c = threadIdx.x;
    const float bx = out[r * 10 + 2], by = out[r * 10 + 3];
    const float bw = out[r * 10 + 4], bh = out[r * 10 + 5];

    int   xs0[6], xs1[6], ys0[6], ys1[6];
    float wxs[6], wys[6];
    #pragma unroll
    for (int s = 0; s < 6; ++s) {
        const float off = (s + 0.5f) / 6.0f;
        float fx = fminf(fmaxf((bx + off * bw) / 32.0f - 0.5f, 0.0f), (float)(MG - 1));
        float fy = fminf(fmaxf((by + off * bh) / 32.0f - 0.5f, 0.0f), (float)(MG - 1));
        const int x0 = (int)floorf(fx);
        const int y0 = (int)floorf(fy);
        xs0[s] = x0; xs1[s] = min(x0 + 1, MG - 1); wxs[s] = fx - (float)x0;
        ys0[s] = y0; ys1[s] = min(y0 + 1, MG - 1); wys[s] = fy - (float)y0;
    }
    #pragma unroll
    for (int ty = 0; ty < 3; ++ty) {
        #pragma unroll
        for (int tx = 0; tx < 3; ++tx) {
            float acc = 0.0f;
            #pragma unroll
            for (int py = 0; py < 2; ++py) {
                #pragma unroll
                for (int px = 0; px < 2; ++px) {
                    const int sy = ty * 2 + py, sx = tx * 2 + px;
                    const float wx = wxs[sx], wy = wys[sy];
                    const float v00 = feat[(ys0[sy] * MG + xs0[sx]) * CCH + c];
                    const float v01 = feat[(ys0[sy] * MG + xs1[sx]) * CCH + c];
                    const float v10 = feat[(ys1[sy] * MG + xs0[sx]) * CCH + c];
                    const float v11 = feat[(ys1[sy] * MG + xs1[sx]) * CCH + c];
                    acc += v00 * (1.0f - wy) * (1.0f - wx) + v01 * (1.0f - wy) * wx
                         + v10 * wy * (1.0f - wx)          + v11 * wy * wx;
                }
            }
            flat[r * FIN + (ty * 3 + tx) * CCH + c] = acc * 0.25f;
        }
    }
}

// ---------------------------------------------------------------------------
// 7) FC1 GEMM (512x2304 @ 2304x1024) + bias + ReLU, LDS-tiled f32 WMMA.
//    Block tile 64x128 staged in LDS per K=32 chunk; 8 waves x (1 Msub, 4 Nsub)
//    16x16 accumulators. A fragment (ds_load_b64) reused across 4 WMMAs.
// ---------------------------------------------------------------------------
__global__ void fc1_wmma_kernel(const float* __restrict__ flat,
                                const float* __restrict__ Wfc1,
                                const float* __restrict__ bfc1,
                                float* __restrict__ h) {
    __shared__ float A_s[BM * ASTR];   // 64 x 32 (+pad)
    __shared__ float B_s[BK * BSTR];   // 32 x 128 (+pad)

    const int tid  = threadIdx.x;
    const int lane = tid & 31;
    const int wave = tid >> 5;
    const int l15  = lane & 15;
    const int half = lane >> 4;        // 0: lanes 0-15 (K,K+1), 1: lanes 16-31 (K+2,K+3)
    const int koff = half * 2;

    const int mblk = blockIdx.x >> 3;            // 8 m-blocks of 64
    const int nblk = blockIdx.x & 7;             // 8 n-blocks of 128
    const int m_block = mblk * BM;
    const int n_block = nblk * BN;

    const int msub = wave & 3;                   // wave's 16-row sub-tile
    const int ngrp = wave >> 2;                  // 0 or 1 -> n offset 0 / 64
    const int arow = (msub * 16 + l15) * ASTR;   // LDS row base for A fragment

    v8f acc[4] = {};

    for (int k0 = 0; k0 < FIN; k0 += BK) {
        __syncthreads();
        // Stage A: 64x32 floats, coalesced (each wave reads a 128B row segment)
        #pragma unroll
        for (int i = 0; i < (BM * BK) / 256; ++i) {
            const int e = tid + i * 256;
            const int r = e >> 5, c = e & 31;
            A_s[r * ASTR + c] = flat[(m_block + r) * FIN + k0 + c];
        }
        // Stage B: 32x128 floats
        #pragma unroll
        for (int i = 0; i < (BK * BN) / 256; ++i) {
            const int e = tid + i * 256;
            const int r = e >> 7, c = e & 127;
            B_s[r * BSTR + c] = Wfc1[(k0 + r) * HID + n_block + c];
        }
        __syncthreads();

        #pragma unroll
        for (int kk = 0; kk < BK; kk += 4) {
            const v2f a = *(const v2f*)&A_s[arow + kk + koff];
            #pragma unroll
            for (int t = 0; t < 4; ++t) {
                const int nb = ngrp * 64 + t * 16 + l15;
                v2f b;
                b.x = B_s[(kk + koff + 0) * BSTR + nb];
                b.y = B_s[(kk + koff + 1) * BSTR + nb];
                acc[t] = __builtin_amdgcn_wmma_f32_16x16x4_f32(
                    false, a, false, b, (short)0, acc[t], false, false);
            }
        }
    }

    // Epilogue: bias + ReLU, D layout: VGPR rr -> M = rr + 8*half, N = l15
    #pragma unroll
    for (int t = 0; t < 4; ++t) {
        const int nIdx = n_block + ngrp * 64 + t * 16 + l15;
        const float bn = bfc1[nIdx];
        #pragma unroll
        for (int rr = 0; rr < 8; ++rr) {
            const int m = m_block + msub * 16 + rr + 8 * half;
            const float v = acc[t][rr] + bn;
            h[m * HID + nIdx] = v > 0.0f ? v : 0.0f;
        }
    }
}

// ---------------------------------------------------------------------------
// 8) FC2 + final unparameterize; one thread per ROI
// ---------------------------------------------------------------------------
__global__ void fc2_final_kernel(const float* __restrict__ h,
                                 const float* __restrict__ Wfc2,
                                 const float* __restrict__ bfc2,
                                 float* __restrict__ out) {
    const int r = blockIdx.x * blockDim.x + threadIdx.x;
    if (r >= RPOST) return;
    float reg[4];
    #pragma unroll
    for (int j = 0; j < 4; ++j) reg[j] = bfc2[j];
    const float* hr = h + r * HID;
    for (int c = 0; c < HID; ++c) {
        const float hv = hr[c];
        #pragma unroll
        for (int j = 0; j < 4; ++j) reg[j] = fmaf(hv, Wfc2[c * 4 + j], reg[j]);
    }
    const float ax = out[r * 10 + 2], ay = out[r * 10 + 3];
    const float aw = out[r * 10 + 4], ah = out[r * 10 + 5];
    const float cxa = ax + aw * 0.5f, cya = ay + ah * 0.5f;
    const float cx = reg[0] * aw + cxa, cy = reg[1] * ah + cya;
    const float w  = aw * expf(reg[2]), hh = ah * expf(reg[3]);
    out[r * 10 + 6] = cx - w * 0.5f;
    out[r * 10 + 7] = cy - hh * 0.5f;
    out[r * 10 + 8] = w;
    out[r * 10 + 9] = hh;
}

// ---------------------------------------------------------------------------
extern "C" void kernel_launch(void* const* d_in, const int* in_sizes, int n_in,
                              void* d_out, int out_size, void* d_ws, size_t ws_size,
                              hipStream_t stream) {
    (void)in_sizes; (void)n_in; (void)out_size; (void)ws_size;
    const float* feat  = (const float*)d_in[0];
    const float* anch  = (const float*)d_in[1];
    const float* Wobj  = (const float*)d_in[2];
    const float* bobj  = (const float*)d_in[3];
    const float* Wreg  = (const float*)d_in[4];
    const float* breg  = (const float*)d_in[5];
    const float* Wfc1  = (const float*)d_in[6];
    const float* bfc1  = (const float*)d_in[7];
    const float* Wfc2  = (const float*)d_in[8];
    const float* bfc2  = (const float*)d_in[9];
    float* out = (float*)d_out;

    float* ws = (float*)d_ws;
    float* scores  = ws;              ws += NA;
    float* boxes   = ws;              ws += NA * 4;
    float* probs   = ws;              ws += NA * 2;
    float* rpn     = ws;              ws += NCELL * 18;
    int*   sidx    = (int*)ws;        ws += NA;
    int*   keep    = (int*)ws;        ws += NA;
    int*   sel     = (int*)ws;        ws += RPOST;
    float* flat    = ws;              ws += RPOST * FIN;
    float* hbuf    = ws;              ws += RPOST * HID;

    rpn_head_kernel<<<NCELL, CCH, 0, stream>>>(feat, Wobj, bobj, Wreg, breg, rpn);
    anchor_decode_kernel<<<NA / 256, 256, 0, stream>>>(rpn, anch, scores, boxes, probs);
    sort_kernel<<<1, 1024, 0, stream>>>(scores, sidx);
    const size_t nms_lds = (size_t)NA * 4 * sizeof(float) + NA + 16;
    nms_kernel<<<1, 256, nms_lds, stream>>>(boxes, sidx, keep);
    select_kernel<<<1, RPOST, 0, stream>>>(sidx, keep, probs, boxes, sel, out);
    roialign_kernel<<<RPOST, CCH, 0, stream>>>(feat, out, flat);
    fc1_wmma_kernel<<<64, 256, 0, stream>>>(flat, Wfc1, bfc1, hbuf);   // 8x8 blocks of 64x128
    fc2_final_kernel<<<2, 256, 0, stream>>>(hbuf, Wfc2, bfc2, out);
}